// GraphConvolution_515396075921
// MI455X (gfx1250) — compile-verified
//
#include <hip/hip_runtime.h>

// GCN layer: out = relu(segment_sum(support[col]*val -> row)),  support = x @ W
// x: [N,512] f32, W: [512,256] f32, edges: row/col int32, val f32.
// Phase 1 uses V_WMMA_F32_16X16X4_F32 (f32-native matrix path on gfx1250).

#define IN_F  512
#define OUT_F 256
#define LDS_PAD 516   // 512 + 4 floats: row stride 516 -> banks (4m + k + {0..3}), conflict-free
#define ROWS_PER_BLOCK 64

typedef float v2f __attribute__((ext_vector_type(2)));
typedef float v8f __attribute__((ext_vector_type(8)));

// ---------------------------------------------------------------------------
// Kernel 1: support = x @ W via WMMA f32 16x16x4.
// Block = 128 threads = 4 waves. Block owns a 64-row strip of x (132 KB LDS).
// Wave w computes a 64x64 output tile: 4 row-tiles x 4 col-tiles of 16x16
// (16 accumulators, 128 acc VGPRs), so each B fragment loaded from L2 feeds
// FOUR wmma ops -> B traffic over the whole GEMM drops to ~0.8 GB.
// A fragments are LDS-broadcast to all 4 waves of the block.
// ---------------------------------------------------------------------------
__global__ __launch_bounds__(128) void gcn_gemm_wmma_f32(
    const float* __restrict__ x, const float* __restrict__ w,
    float* __restrict__ support, int n_nodes)
{
    __shared__ float xs[ROWS_PER_BLOCK * LDS_PAD];   // ~132 KB

    const int tid  = threadIdx.x;
    const int row0 = blockIdx.x * ROWS_PER_BLOCK;

    // Cooperative float4 load of the 64x512 strip of x into LDS.
    // 128 float4 per row, 128 threads: thread t loads float4 slot t of row r.
    #pragma unroll 4
    for (int r = 0; r < ROWS_PER_BLOCK; ++r) {
        float4 v = make_float4(0.f, 0.f, 0.f, 0.f);
        int gr = row0 + r;
        if (gr < n_nodes)
            v = *(const float4*)(x + (size_t)gr * IN_F + tid * 4);
        *(float4*)(xs + r * LDS_PAD + tid * 4) = v;
    }
    __syncthreads();

    const int lane  = tid & 31;
    const int wave  = tid >> 5;
    const int m16   = lane & 15;        // row within tile (A) / col within tile (B,C)
    const int khalf = (lane >> 4) * 2;  // lanes 0-15: K=0,1 ; lanes 16-31: K=2,3
    const int n0    = wave * 64;

    // 16 accumulators: [row-tile 0..3] x [col-tile 0..3]
    v8f acc[4][4];
    #pragma unroll
    for (int rt = 0; rt < 4; ++rt)
        #pragma unroll
        for (int ct = 0; ct < 4; ++ct)
            acc[rt][ct] = (v8f){};

    #pragma unroll 2
    for (int k = 0; k < IN_F; k += 4) {
        // A fragments (16x4 f32 each): ds_load_b64, conflict-free via LDS_PAD.
        v2f a[4];
        #pragma unroll
        for (int rt = 0; rt < 4; ++rt) {
            const float* xr = xs + (rt * 16 + m16) * LDS_PAD + khalf + k;
            a[rt].x = xr[0];
            a[rt].y = xr[1];
        }

        // B fragments (4x16 f32), K rows strided by OUT_F in global; W is
        // 512 KB and stays resident in the 192 MB L2 across all blocks.
        const float* wr0 = w + (size_t)(k + khalf) * OUT_F + n0 + m16;
        const float* wr1 = wr0 + OUT_F;
        v2f b[4];
        #pragma unroll
        for (int ct = 0; ct < 4; ++ct) {
            b[ct].x = wr0[ct * 16];
            b[ct].y = wr1[ct * 16];
        }

        // 16 independent WMMAs: each B fragment feeds 4, each A fragment feeds 4.
        #pragma unroll
        for (int ct = 0; ct < 4; ++ct)
            #pragma unroll
            for (int rt = 0; rt < 4; ++rt)
                acc[rt][ct] = __builtin_amdgcn_wmma_f32_16x16x4_f32(
                    false, a[rt], false, b[ct], (short)0, acc[rt][ct],
                    false, false);
    }

    // Store C/D per 16x16 f32 layout: VGPR r, lanes 0-15 -> M=r, lanes 16-31 -> M=r+8.
    const int mbase = (lane >> 4) * 8;
    #pragma unroll
    for (int rt = 0; rt < 4; ++rt) {
        #pragma unroll
        for (int r = 0; r < 8; ++r) {
            int grow = row0 + rt * 16 + mbase + r;
            if (grow < n_nodes) {
                float* orow = support + (size_t)grow * OUT_F + n0 + m16;
                orow[0]  = acc[rt][0][r];
                orow[16] = acc[rt][1][r];
                orow[32] = acc[rt][2][r];
                orow[48] = acc[rt][3][r];
            }
        }
    }
}

// ---------------------------------------------------------------------------
// Kernel 2: edge scatter.  64 lanes per edge, float4 per lane (256 floats).
// support gathers hit L2 (102 MB resident); scatter uses HW f32 atomics.
// ---------------------------------------------------------------------------
__global__ __launch_bounds__(256) void gcn_spmm_scatter(
    const float* __restrict__ support, const int* __restrict__ erow,
    const int* __restrict__ ecol, const float* __restrict__ eval,
    float* __restrict__ out, int n_edges)
{
    const int e = blockIdx.x * 4 + (threadIdx.x >> 6);
    const int t = threadIdx.x & 63;
    if (e >= n_edges) return;

    const int   r = erow[e];
    const int   c = ecol[e];
    const float v = eval[e];

    const float4 s = *(const float4*)(support + (size_t)c * OUT_F + t * 4);
    float* o = out + (size_t)r * OUT_F + t * 4;

    __hip_atomic_fetch_add(o + 0, s.x * v, __ATOMIC_RELAXED, __HIP_MEMORY_SCOPE_AGENT);
    __hip_atomic_fetch_add(o + 1, s.y * v, __ATOMIC_RELAXED, __HIP_MEMORY_SCOPE_AGENT);
    __hip_atomic_fetch_add(o + 2, s.z * v, __ATOMIC_RELAXED, __HIP_MEMORY_SCOPE_AGENT);
    __hip_atomic_fetch_add(o + 3, s.w * v, __ATOMIC_RELAXED, __HIP_MEMORY_SCOPE_AGENT);
}

// ---------------------------------------------------------------------------
// Kernel 3: in-place ReLU, float4-vectorized.
// ---------------------------------------------------------------------------
__global__ __launch_bounds__(256) void gcn_relu(float* __restrict__ out, int n4)
{
    int i = blockIdx.x * 256 + threadIdx.x;
    if (i >= n4) return;
    float4 v = ((float4*)out)[i];
    v.x = fmaxf(v.x, 0.f); v.y = fmaxf(v.y, 0.f);
    v.z = fmaxf(v.z, 0.f); v.w = fmaxf(v.w, 0.f);
    ((float4*)out)[i] = v;
}

// ---------------------------------------------------------------------------
extern "C" void kernel_launch(void* const* d_in, const int* in_sizes, int n_in,
                              void* d_out, int out_size, void* d_ws, size_t ws_size,
                              hipStream_t stream)
{
    const float* x    = (const float*)d_in[0];
    const float* w    = (const float*)d_in[1];
    const int*   erow = (const int*)d_in[2];
    const int*   ecol = (const int*)d_in[3];
    const float* eval = (const float*)d_in[4];
    float* out = (float*)d_out;

    const int n_nodes = in_sizes[0] / IN_F;
    const int n_edges = in_sizes[2];

    float* support = (float*)d_ws;   // n_nodes * OUT_F floats of scratch

    // out accumulates edge contributions -> must start at zero each call.
    hipMemsetAsync(d_out, 0, (size_t)out_size * sizeof(float), stream);

    // 1) support = x @ W  (WMMA f32, 64 rows per block)
    const int strips = (n_nodes + ROWS_PER_BLOCK - 1) / ROWS_PER_BLOCK;
    gcn_gemm_wmma_f32<<<strips, 128, 0, stream>>>(x, w, support, n_nodes);

    // 2) scatter-add over edges
    const int eblocks = (n_edges + 3) / 4;
    gcn_spmm_scatter<<<eblocks, 256, 0, stream>>>(support, erow, ecol, eval,
                                                  out, n_edges);

    // 3) relu
    const int n4 = out_size / 4;
    gcn_relu<<<(n4 + 255) / 256, 256, 0, stream>>>(out, n4);
}